// Roundings_24154896073503
// MI455X (gfx1250) — compile-verified
//
#include <hip/hip_runtime.h>

// ---------------------------------------------------------------------------
// Soft-rounding staircase: out = sum_{k=0..30} sigmoid(100*(x+15.5-k)) - 16
//
// Sharpness 100 => in f32 ONLY the nearest staircase term is fractional:
// sigmoid(+-50) = 1 -+ 2e-22 rounds to exactly 1.0f / 0.0f. So with
//   t = x + 15.5, kcl = med3(rndne(t), 0, 30):
//   out = (kcl - 16) + 1/(1 + 2^(144.2695*(kcl - t)))
// (~10 VALU, 2 transcendental per element -> memory bound at 23.3 TB/s).
//
// Data movement: CDNA5 async global->LDS (ASYNCcnt), per-wave double
// buffering (2x b128 issues per 1KB buffer), ds_load_b128 + s_wait_dscnt
// to consume, non-temporal b128 stores out (268 MB stream > 192 MB L2).
// ---------------------------------------------------------------------------

typedef float vfloat4 __attribute__((ext_vector_type(4)));

#define WAVE_SZ     32
#define WAVES_PB    8
#define BLOCK_SZ    (WAVE_SZ * WAVES_PB)      // 256 threads = 8 waves
#define F4_PER_LANE 2                         // float4s per lane per chunk
#define CHUNK_F4    (WAVE_SZ * F4_PER_LANE)   // 64 float4 per wave-chunk
#define BUF_BYTES   (CHUNK_F4 * 16)           // 1 KB per buffer
#define HALF_BUF    (WAVE_SZ * 16)            // 512 B
#define LDS_PER_WV  (2 * BUF_BYTES)           // double buffer
#define LDS_BYTES   (WAVES_PB * LDS_PER_WV)   // 16 KB per block (dynamic LDS)
#define GRID_BLOCKS 2048

__device__ __forceinline__ void async_copy_b128(unsigned lds_addr, const vfloat4* gptr) {
  // GLOBAL_LOAD_ASYNC_TO_LDS_B128, GV mode; tracked by ASYNCcnt.
  asm volatile("global_load_async_to_lds_b128 %0, %1, off"
               :: "v"(lds_addr), "v"(gptr)
               : "memory");
}

__device__ __forceinline__ void lds_read_2x_b128(unsigned a0, unsigned a1,
                                                 vfloat4& v0, vfloat4& v1) {
  asm volatile("ds_load_b128 %0, %2\n\t"
               "ds_load_b128 %1, %3\n\t"
               "s_wait_dscnt 0x0"
               : "=v"(v0), "=v"(v1)
               : "v"(a0), "v"(a1)
               : "memory");
}

__device__ __forceinline__ float soft_round(float x) {
  const float t   = x + 15.5f;                              // d_k = t - k
  const float kf  = __builtin_rintf(t);                     // v_rndne_f32
  const float kcl = __builtin_amdgcn_fmed3f(kf, 0.0f, 30.0f); // v_med3_f32
  // sigmoid(100*(t-kcl)) = 1 / (1 + 2^(100*log2(e)*(kcl-t)))
  const float e = __builtin_amdgcn_exp2f((kcl - t) * 144.26950408889634f);
  const float s = __builtin_amdgcn_rcpf(1.0f + e);
  return (kcl - 16.0f) + s;
}

__global__ void __launch_bounds__(BLOCK_SZ)
roundings_async_kernel(const float* __restrict__ xin,
                       float* __restrict__ out,
                       long long n) {
  const long long n4 = n >> 2;
  const int lane = threadIdx.x & (WAVE_SZ - 1);
  const int wv   = threadIdx.x >> 5;                       // wave32
  const long long waveId     = (long long)blockIdx.x * WAVES_PB + wv;
  const long long waveStride = (long long)gridDim.x * WAVES_PB;
  const long long nChunks    = (n4 + CHUNK_F4 - 1) / CHUNK_F4;

  const vfloat4* in4  = (const vfloat4*)xin;
  vfloat4*       out4 = (vfloat4*)out;

  // Raw LDS byte offsets within the dynamic LDS region; the async writes and
  // ds_load reads use the same offsets, so placement is self-consistent.
  const unsigned ldsBase = (unsigned)(wv * LDS_PER_WV);
  const unsigned slotA0 = ldsBase + (unsigned)lane * 16u;
  const unsigned slotA1 = slotA0 + (unsigned)HALF_BUF;
  const unsigned slotB0 = slotA0 + (unsigned)BUF_BYTES;
  const unsigned slotB1 = slotB0 + (unsigned)HALF_BUF;

  // Scalar tail (n % 4): wave 0 of block 0, straight from global.
  if (blockIdx.x == 0 && wv == 0) {
    const long long tail = n4 << 2;
    if (lane < (int)(n - tail)) out[tail + lane] = soft_round(xin[tail + lane]);
  }

  long long c = waveId;
  if (c >= nChunks) return;

  // Prologue: prefetch first chunk into buffer A (up to 2 async b128 issues).
  {
    const long long i0 = c * CHUNK_F4 + lane;
    const long long i1 = i0 + WAVE_SZ;
    if (i0 < n4) async_copy_b128(slotA0, in4 + i0);
    if (i1 < n4) async_copy_b128(slotA1, in4 + i1);
  }

  int cur = 0;
  for (;;) {
    const long long nextC = c + waveStride;
    const bool hasNext = (nextC < nChunks);
    if (hasNext) {
      // Prefetch chunk N+1 into the other buffer, then drain everything
      // except the (<=2) newest async ops: chunk N has landed in LDS.
      const long long i0 = nextC * CHUNK_F4 + lane;
      const long long i1 = i0 + WAVE_SZ;
      if (i0 < n4) async_copy_b128(cur ? slotA0 : slotB0, in4 + i0);
      if (i1 < n4) async_copy_b128(cur ? slotA1 : slotB1, in4 + i1);
      asm volatile("s_wait_asynccnt 0x2" ::: "memory");
    } else {
      asm volatile("s_wait_asynccnt 0x0" ::: "memory");
    }

    const long long i0 = c * CHUNK_F4 + lane;
    const long long i1 = i0 + WAVE_SZ;
    vfloat4 v0, v1;
    lds_read_2x_b128(cur ? slotB0 : slotA0, cur ? slotB1 : slotA1, v0, v1);
    if (i0 < n4) {
      vfloat4 r;
      r.x = soft_round(v0.x); r.y = soft_round(v0.y);
      r.z = soft_round(v0.z); r.w = soft_round(v0.w);
      __builtin_nontemporal_store(r, out4 + i0);
    }
    if (i1 < n4) {
      vfloat4 r;
      r.x = soft_round(v1.x); r.y = soft_round(v1.y);
      r.z = soft_round(v1.z); r.w = soft_round(v1.w);
      __builtin_nontemporal_store(r, out4 + i1);
    }

    if (!hasNext) break;
    c = nextC;
    cur ^= 1;
  }
}

extern "C" void kernel_launch(void* const* d_in, const int* in_sizes, int n_in,
                              void* d_out, int out_size, void* d_ws, size_t ws_size,
                              hipStream_t stream) {
  (void)n_in; (void)d_ws; (void)ws_size; (void)out_size;
  const float* x = (const float*)d_in[0];
  float* out = (float*)d_out;
  const long long n = (long long)in_sizes[0];

  const long long nChunks = ((n >> 2) + CHUNK_F4 - 1) / CHUNK_F4;
  const long long blocksNeeded = (nChunks + WAVES_PB - 1) / WAVES_PB;
  const int grid = (int)(blocksNeeded < GRID_BLOCKS
                             ? (blocksNeeded > 0 ? blocksNeeded : 1)
                             : GRID_BLOCKS);

  roundings_async_kernel<<<grid, BLOCK_SZ, LDS_BYTES, stream>>>(x, out, n);
}